// HModel_59648505807204
// MI455X (gfx1250) — compile-verified
//
// CDNA5 / gfx1250 implementation of the HModel GNN (orbital block + seq block).
//
// Strategy: dense GEMMs are ~40 GFLOP total -> trivial vs the WMMA ceiling;
// the workload is bound by edge gather/scatter traffic (~6-8 GB) and float
// atomics (HBM 23.3 TB/s -> ~0.3-0.5 ms floor). All linear layers use
// V_WMMA_F32_16X16X4_F32 (exact fp32; no reason to quantize a memory-bound
// graph net). The TransformerConv edge-linear is factored algebraically
// (qe / eagg tricks) to avoid materializing E x 256 intermediates
// (saves ~2.5 GB of traffic per tconv, x3 tconvs).
//
// Param leaf order assumes jax pytree flattening (dict keys sorted):
//   0:x 1:edge_attr
//   orb.gat: 2:att 3:bias 4:e.W 5:e.b 6:l.W 7:l.b 8:r.W 9:r.b
//   orb.gine1: 10:eps 11:lin_e.W 12:lin_e.b 13:mlp1.W 14:mlp1.b 15:mlp2.W 16:mlp2.b
//   orb.gine2: 17..23 (same layout, hid=32)
//   orb.t:     24:e.W 25:e.b 26:k.W 27:k.b 28:q.W 29:q.b 30:skip.W 31:skip.b 32:v.W 33:v.b
//   seq.gat:   34..41   seq.gine: 42..48
//   seq.t1:    49..58   seq.t2:   59..68
//   69:w  70:edge_index (int32, [2,E] row-major: src then dst)

#include <hip/hip_runtime.h>

typedef float v2f __attribute__((ext_vector_type(2)));
typedef float v8f __attribute__((ext_vector_type(8)));

#define NN 50000
#define NE 800000

// ---------------------------------------------------------------------------
// ordered-float atomic max (init target to -inf bit pattern 0xFF800000)
__device__ __forceinline__ void atomicMaxF(float* a, float v) {
  if (v >= 0.0f) atomicMax((int*)a, __float_as_int(v));
  else           atomicMin((unsigned int*)a, __float_as_uint(v));
}

// ---------------------------------------------------------------------------
__global__ void fill_kernel(float* __restrict__ p, int n, float v) {
  int i = blockIdx.x * 256 + threadIdx.x;
  if (i < n) p[i] = v;
}

// ---------------------------------------------------------------------------
// Generic fp32 WMMA GEMM:  C[M x Nc] = act(A[M x K] @ B + bias) (+C if accum)
// One wave computes a 16x32 tile (two independent 16x16 accumulator chains
// sharing one A fragment -> WMMA latency hiding, branch-free inner loop).
// grid.x = M/16, grid.y = Nc/32.  TRANSB: B stored [Nc x K] row-major.
template <int TRANSB>
__global__ void gemm_wmma(const float* __restrict__ A, int lda,
                          const float* __restrict__ B, int ldb,
                          const float* __restrict__ bias,
                          float* __restrict__ C, int ldc,
                          int K, int relu, int accum) {
  int lane = threadIdx.x;          // block = 32 threads = 1 wave
  int m0 = blockIdx.x * 16;
  int n0 = blockIdx.y * 32;
  int half = lane >> 4;            // lanes 16..31: K+2 in frags, rows M+8 in C
  int l16  = lane & 15;

  const float* ap = A + (size_t)(m0 + l16) * lda + half * 2;
  const float* bp0;
  const float* bp1;
  int bstep;
  if (TRANSB) {                    // B[n][k]
    bp0 = B + (size_t)(n0 + l16) * ldb + half * 2;
    bp1 = bp0 + (size_t)16 * ldb;
    bstep = 4;
  } else {                         // B[k][n]
    bp0 = B + (size_t)(half * 2) * ldb + n0 + l16;
    bp1 = bp0 + 16;
    bstep = 4 * ldb;
  }

  v8f acc0 = {}, acc1 = {};
  for (int k = 0; k < K; k += 4) {
    v2f a = *(const v2f*)ap;
    v2f b0, b1;
    if (TRANSB) {
      b0 = *(const v2f*)bp0;
      b1 = *(const v2f*)bp1;
    } else {
      b0.x = bp0[0]; b0.y = bp0[ldb];
      b1.x = bp1[0]; b1.y = bp1[ldb];
    }
    acc0 = __builtin_amdgcn_wmma_f32_16x16x4_f32(false, a, false, b0,
                                                 (short)0, acc0, false, false);
    acc1 = __builtin_amdgcn_wmma_f32_16x16x4_f32(false, a, false, b1,
                                                 (short)0, acc1, false, false);
    ap += 4; bp0 += bstep; bp1 += bstep;
  }

  int col0 = n0 + l16, col1 = col0 + 16;
  float bv0 = bias ? bias[col0] : 0.0f;
  float bv1 = bias ? bias[col1] : 0.0f;
#pragma unroll
  for (int r = 0; r < 8; ++r) {
    int rr = m0 + r + half * 8;                 // C frag: VGPR r -> row r (+8)
    size_t i0 = (size_t)rr * ldc + col0;
    size_t i1 = (size_t)rr * ldc + col1;
    float o0 = acc0[r] + bv0;
    float o1 = acc1[r] + bv1;
    if (accum) { o0 += C[i0]; o1 += C[i1]; }
    if (relu)  { o0 = fmaxf(o0, 0.0f); o1 = fmaxf(o1, 0.0f); }
    C[i0] = o0;
    C[i1] = o1;
  }
}

// ---------------------------------------------------------------------------
// GINE: agg[dst] += relu(x[src] + lin_e(ea));  one wave per edge (64 cols)
__global__ void gine_scatter(const float* __restrict__ x,
                             const float* __restrict__ elin,
                             const int* __restrict__ src,
                             const int* __restrict__ dst,
                             float* __restrict__ agg) {
  int wid = threadIdx.x >> 5, lane = threadIdx.x & 31;
  int e = blockIdx.x * 8 + wid;
  if (e >= NE) return;
  int s = src[e], d = dst[e];
#pragma unroll
  for (int t = 0; t < 2; ++t) {
    int c = lane + t * 32;
    float v = x[(size_t)s * 64 + c] + elin[(size_t)e * 64 + c];
    atomicAdd(&agg[(size_t)d * 64 + c], fmaxf(v, 0.0f));
  }
}

__global__ void gine_update(const float* __restrict__ x,
                            const float* __restrict__ agg,
                            const float* __restrict__ eps,
                            float* __restrict__ out, int n) {
  int i = blockIdx.x * 256 + threadIdx.x;
  if (i < n) out[i] = (1.0f + eps[0]) * x[i] + agg[i];
}

// ---------------------------------------------------------------------------
// TransformerConv.  logits = (q[dst].k[src] + ea.qe[dst] + qb[dst]) / 8
// qb[n,h] = sum_d q[n,h,d] * be[h,d]   (node-level bias fold)
__global__ void tconv_qb(const float* __restrict__ q,
                         const float* __restrict__ be,
                         float* __restrict__ qb) {
  int i = blockIdx.x * 256 + threadIdx.x;     // i = n*4 + h
  if (i >= NN * 4) return;
  int n = i >> 2, h = i & 3;
  float s = 0.0f;
  for (int d = 0; d < 64; ++d)
    s += q[(size_t)n * 256 + h * 64 + d] * be[h * 64 + d];
  qb[i] = s;
}

// pass1: wave per edge; lane = h*8+j; 8-lane shfl reduction per head
__global__ void tconv_pass1(const float* __restrict__ q,
                            const float* __restrict__ k,
                            const float* __restrict__ qe,
                            const float* __restrict__ qb,
                            const float* __restrict__ ea,
                            const int* __restrict__ src,
                            const int* __restrict__ dst,
                            float* __restrict__ logits,   // E x 4
                            float* __restrict__ mmax) {   // N x 4
  int wid = threadIdx.x >> 5, lane = threadIdx.x & 31;
  int e = blockIdx.x * 8 + wid;
  if (e >= NE) return;
  int s = src[e], d = dst[e];
  int h = lane >> 3, j = lane & 7;
  const float* qr = q  + (size_t)d * 256 + h * 64;
  const float* kr = k  + (size_t)s * 256 + h * 64;
  const float* qn = qe + (size_t)d * 128 + h * 32;
  const float* er = ea + (size_t)e * 32;
  float p = 0.0f;
#pragma unroll
  for (int t = 0; t < 8; ++t) p += qr[j * 8 + t] * kr[j * 8 + t];
#pragma unroll
  for (int t = 0; t < 4; ++t) p += qn[j * 4 + t] * er[j * 4 + t];
  p += __shfl_xor(p, 1, 32);
  p += __shfl_xor(p, 2, 32);
  p += __shfl_xor(p, 4, 32);
  if (j == 0) {
    float lg = (p + qb[d * 4 + h]) * 0.125f;   // / sqrt(64)
    logits[(size_t)e * 4 + h] = lg;
    atomicMaxF(&mmax[d * 4 + h], lg);
  }
}

__global__ void tconv_pass2(const int* __restrict__ dst,
                            const float* __restrict__ mmax,
                            float* __restrict__ logits,
                            float* __restrict__ ssum) {
  int i = blockIdx.x * 256 + threadIdx.x;     // i = e*4 + h
  if (i >= NE * 4) return;
  int e = i >> 2, h = i & 3;
  int d = dst[e];
  float ex = expf(logits[i] - mmax[d * 4 + h]);
  logits[i] = ex;
  atomicAdd(&ssum[d * 4 + h], ex);
}

// pass3: accum[dst] += ex_h * v[src]; eagg[dst,h,:] += ex_h * ea[e,:]
__global__ void tconv_pass3(const float* __restrict__ v,
                            const float* __restrict__ ea,
                            const float* __restrict__ exl,
                            const int* __restrict__ src,
                            const int* __restrict__ dst,
                            float* __restrict__ accum,    // N x 256
                            float* __restrict__ eagg) {   // N x 128
  int wid = threadIdx.x >> 5, lane = threadIdx.x & 31;
  int e = blockIdx.x * 8 + wid;
  if (e >= NE) return;
  int s = src[e], d = dst[e];
  float exh[4];
#pragma unroll
  for (int h = 0; h < 4; ++h) exh[h] = exl[(size_t)e * 4 + h];
#pragma unroll
  for (int t = 0; t < 8; ++t) {
    int c = lane + t * 32;
    atomicAdd(&accum[(size_t)d * 256 + c], exh[c >> 6] * v[(size_t)s * 256 + c]);
  }
  float eav = ea[(size_t)e * 32 + lane];
#pragma unroll
  for (int t = 0; t < 4; ++t)
    atomicAdd(&eagg[(size_t)d * 128 + t * 32 + lane], exh[t] * eav);
}

__global__ void tconv_final_concat(const float* __restrict__ accum,
                                   const float* __restrict__ ssum,
                                   const float* __restrict__ be,
                                   const float* __restrict__ skip,
                                   float* __restrict__ out) {
  int i = blockIdx.x * 256 + threadIdx.x;     // n*256 + c
  if (i >= NN * 256) return;
  int node = i >> 8, c = i & 255, h = c >> 6;
  float s = ssum[node * 4 + h];
  out[i] = (accum[i] + s * be[c]) / (s + 1e-16f) + skip[i];
}

__global__ void tconv_final_mean(const float* __restrict__ accum,
                                 const float* __restrict__ ssum,
                                 const float* __restrict__ be,
                                 const float* __restrict__ skip,
                                 float* __restrict__ out) {
  int i = blockIdx.x * 256 + threadIdx.x;     // n*64 + d
  if (i >= NN * 64) return;
  int node = i >> 6, dd = i & 63;
  float r = 0.0f;
#pragma unroll
  for (int h = 0; h < 4; ++h) {
    float s = ssum[node * 4 + h];
    r += (accum[(size_t)node * 256 + h * 64 + dd] + s * be[h * 64 + dd]) /
         (s + 1e-16f);
  }
  out[i] = r * 0.25f + skip[i];
}

// ---------------------------------------------------------------------------
// GATv2
__global__ void gat_pass1(const float* __restrict__ xl,
                          const float* __restrict__ xr,
                          const float* __restrict__ ee,
                          const float* __restrict__ att,
                          const int* __restrict__ src,
                          const int* __restrict__ dst,
                          float* __restrict__ logit,
                          float* __restrict__ mmax) {
  int wid = threadIdx.x >> 5, lane = threadIdx.x & 31;
  int e = blockIdx.x * 8 + wid;
  if (e >= NE) return;
  int s = src[e], d = dst[e];
  float p = 0.0f;
#pragma unroll
  for (int t = 0; t < 2; ++t) {
    int c = lane + t * 32;
    float z = xl[(size_t)s * 64 + c] + xr[(size_t)d * 64 + c] +
              ee[(size_t)e * 64 + c];
    z = (z >= 0.0f) ? z : 0.2f * z;            // leaky_relu(0.2)
    p += z * att[c];
  }
  p += __shfl_xor(p, 1, 32);
  p += __shfl_xor(p, 2, 32);
  p += __shfl_xor(p, 4, 32);
  p += __shfl_xor(p, 8, 32);
  p += __shfl_xor(p, 16, 32);
  if (lane == 0) { logit[e] = p; atomicMaxF(&mmax[d], p); }
}

__global__ void gat_pass2(const int* __restrict__ dst,
                          const float* __restrict__ mmax,
                          float* __restrict__ logit,
                          float* __restrict__ ssum) {
  int e = blockIdx.x * 256 + threadIdx.x;
  if (e >= NE) return;
  float ex = expf(logit[e] - mmax[dst[e]]);
  logit[e] = ex;
  atomicAdd(&ssum[dst[e]], ex);
}

__global__ void gat_pass3(const float* __restrict__ xl,
                          const float* __restrict__ exv,
                          const int* __restrict__ src,
                          const int* __restrict__ dst,
                          float* __restrict__ accb) {
  int wid = threadIdx.x >> 5, lane = threadIdx.x & 31;
  int e = blockIdx.x * 8 + wid;
  if (e >= NE) return;
  int s = src[e], d = dst[e];
  float ex = exv[e];
#pragma unroll
  for (int t = 0; t < 2; ++t) {
    int c = lane + t * 32;
    atomicAdd(&accb[(size_t)d * 64 + c], ex * xl[(size_t)s * 64 + c]);
  }
}

__global__ void gat_final(const float* __restrict__ accb,
                          const float* __restrict__ ssum,
                          const float* __restrict__ bias,
                          float* __restrict__ out) {
  int i = blockIdx.x * 256 + threadIdx.x;
  if (i >= NN * 64) return;
  int node = i >> 6, c = i & 63;
  out[i] = accb[i] / (ssum[node] + 1e-16f) + bias[c];
}

// ---------------------------------------------------------------------------
// final merge: x_out = x_orb + s*wsum ; e_out = ea*wsum
__global__ void merge_kernel(const float* __restrict__ xo,
                             const float* __restrict__ sout,
                             const float* __restrict__ ea,
                             const float* __restrict__ w,
                             float* __restrict__ out) {
  int i = blockIdx.x * 256 + threadIdx.x;
  float wsum = (w[0] + w[1] + w[2]) * (1.0f / 3.0f);
  const int nx = NN * 64;
  if (i < nx)                 out[i] = xo[i] + sout[i] * wsum;
  else if (i < nx + NE * 32)  out[i] = ea[i - nx] * wsum;
}

// ---------------------------------------------------------------------------
extern "C" void kernel_launch(void* const* d_in, const int* in_sizes, int n_in,
                              void* d_out, int out_size, void* d_ws,
                              size_t ws_size, hipStream_t stream) {
  (void)in_sizes; (void)n_in; (void)out_size; (void)ws_size;
  const float* x   = (const float*)d_in[0];
  const float* ea  = (const float*)d_in[1];
  const float* w3  = (const float*)d_in[69];
  const int*   ei  = (const int*)d_in[70];
  const int*   src = ei;
  const int*   dst = ei + NE;
  auto P = [&](int i) { return (const float*)d_in[i]; };

  // ---- workspace arena (float, 256B-aligned chunks) ----
  float* base = (float*)d_ws;
  size_t off = 0;
  auto alloc = [&](size_t n) { float* p = base + off; off += (n + 63) & ~(size_t)63; return p; };
  float* xo    = alloc((size_t)NN * 64);
  float* qb_   = alloc((size_t)NN * 256);   // q
  float* kb_   = alloc((size_t)NN * 256);   // k / tconv-out
  float* vb_   = alloc((size_t)NN * 256);   // v
  float* accum = alloc((size_t)NN * 256);
  float* qe    = alloc((size_t)NN * 128);   // also gine2 hidden
  float* eagg  = alloc((size_t)NN * 128);
  float* nA    = alloc((size_t)NN * 64);
  float* nB    = alloc((size_t)NN * 64);
  float* nC    = alloc((size_t)NN * 64);
  float* mN    = alloc((size_t)NN * 4);
  float* sN    = alloc((size_t)NN * 4);
  float* qbv   = alloc((size_t)NN * 4);
  float* e64   = alloc((size_t)NE * 64);
  float* e4    = alloc((size_t)NE * 4);

  const float NEGINF = -__builtin_huge_valf();

  auto fill = [&](float* p, size_t n, float v) {
    fill_kernel<<<dim3((unsigned)((n + 255) / 256)), 256, 0, stream>>>(p, (int)n, v);
  };
  auto gemm = [&](const float* A, int lda, const float* B, int ldb, int tB,
                  const float* bias, float* C, int ldc, int M, int K, int Nc,
                  int relu, int acc) {
    dim3 g(M / 16, Nc / 32);
    if (tB)
      gemm_wmma<1><<<g, 32, 0, stream>>>(A, lda, B, ldb, bias, C, ldc, K, relu, acc);
    else
      gemm_wmma<0><<<g, 32, 0, stream>>>(A, lda, B, ldb, bias, C, ldc, K, relu, acc);
  };

  // ---- GINEConv: ip -> eps, lin_e.W, lin_e.b, mlp1.W, mlp1.b, mlp2.W, mlp2.b
  auto gine = [&](const float* xin, int ip, float* agg, float* upd,
                  float* hid, int hdim, float* hout) {
    gemm(ea, 32, P(ip + 1), 64, 0, P(ip + 2), e64, 64, NE, 32, 64, 0, 0);
    fill(agg, (size_t)NN * 64, 0.0f);
    gine_scatter<<<NE / 8, 256, 0, stream>>>(xin, e64, src, dst, agg);
    gine_update<<<(NN * 64 + 255) / 256, 256, 0, stream>>>(xin, agg, P(ip), upd, NN * 64);
    gemm(upd, 64, P(ip + 3), hdim, 0, P(ip + 4), hid, hdim, NN, 64, hdim, 1, 0);
    gemm(hid, hdim, P(ip + 5), 64, 0, P(ip + 6), hout, 64, NN, hdim, 64, 0, 0);
  };

  // ---- TransformerConv: ip -> e.W,e.b,k.W,k.b,q.W,q.b,skip.W,skip.b,v.W,v.b
  auto tconv = [&](const float* xin, int ip, int concat, float* skipbuf, float* outp) {
    gemm(xin, 64, P(ip + 4), 256, 0, P(ip + 5), qb_, 256, NN, 64, 256, 0, 0); // q
    gemm(xin, 64, P(ip + 2), 256, 0, P(ip + 3), kb_, 256, NN, 64, 256, 0, 0); // k
    gemm(xin, 64, P(ip + 8), 256, 0, P(ip + 9), vb_, 256, NN, 64, 256, 0, 0); // v
    for (int h = 0; h < 4; ++h)      // qe[:,h,:] = q[:,h,:] @ We[:,h,:]^T
      gemm(qb_ + h * 64, 256, P(ip) + h * 64, 256, 1, nullptr,
           qe + h * 32, 128, NN, 64, 32, 0, 0);
    tconv_qb<<<(NN * 4 + 255) / 256, 256, 0, stream>>>(qb_, P(ip + 1), qbv);
    fill(mN, (size_t)NN * 4, NEGINF);
    fill(sN, (size_t)NN * 4, 0.0f);
    fill(accum, (size_t)NN * 256, 0.0f);
    fill(eagg, (size_t)NN * 128, 0.0f);
    tconv_pass1<<<NE / 8, 256, 0, stream>>>(qb_, kb_, qe, qbv, ea, src, dst, e4, mN);
    tconv_pass2<<<(NE * 4 + 255) / 256, 256, 0, stream>>>(dst, mN, e4, sN);
    tconv_pass3<<<NE / 8, 256, 0, stream>>>(vb_, ea, e4, src, dst, accum, eagg);
    for (int h = 0; h < 4; ++h)      // accum[:,h,:] += eagg[:,h,:] @ We[:,h,:]
      gemm(eagg + h * 32, 128, P(ip) + h * 64, 256, 0, nullptr,
           accum + h * 64, 256, NN, 32, 64, 0, 1);
    int sn = concat ? 256 : 64;
    gemm(xin, 64, P(ip + 6), sn, 0, P(ip + 7), skipbuf, sn, NN, 64, sn, 0, 0);
    if (concat)
      tconv_final_concat<<<(NN * 256 + 255) / 256, 256, 0, stream>>>(
          accum, sN, P(ip + 1), skipbuf, outp);
    else
      tconv_final_mean<<<(NN * 64 + 255) / 256, 256, 0, stream>>>(
          accum, sN, P(ip + 1), skipbuf, outp);
  };

  // ---- GATv2: ip -> att, bias, e.W, e.b, l.W, l.b, r.W, r.b  (din=256)
  auto gat = [&](const float* xin, int ip, float* xl, float* xr, float* accb,
                 float* outp) {
    gemm(xin, 256, P(ip + 4), 64, 0, P(ip + 5), xl, 64, NN, 256, 64, 0, 0);
    gemm(xin, 256, P(ip + 6), 64, 0, P(ip + 7), xr, 64, NN, 256, 64, 0, 0);
    gemm(ea, 32, P(ip + 2), 64, 0, P(ip + 3), e64, 64, NE, 32, 64, 0, 0);
    fill(mN, (size_t)NN, NEGINF);
    fill(sN, (size_t)NN, 0.0f);
    fill(accb, (size_t)NN * 64, 0.0f);
    gat_pass1<<<NE / 8, 256, 0, stream>>>(xl, xr, e64, P(ip), src, dst, e4, mN);
    gat_pass2<<<(NE + 255) / 256, 256, 0, stream>>>(dst, mN, e4, sN);
    gat_pass3<<<NE / 8, 256, 0, stream>>>(xl, e4, src, dst, accb);
    gat_final<<<(NN * 64 + 255) / 256, 256, 0, stream>>>(accb, sN, P(ip + 1), outp);
  };

  // ================= orbital block =================
  gine(x,  10, nA, nB, nC, 64, nA);        // gine1 -> nA
  gine(nA, 17, nB, nC, qe, 32, nA);        // gine2 -> nA
  tconv(nA, 24, 1, qb_, kb_);              // tconv (concat) -> kb_ [N x 256]
  gat(kb_, 2, nA, nB, nC, xo);             // gatv2 -> xo = x_orb

  // ================= seq block =====================
  gine(xo, 42, nA, nB, nC, 64, nA);        // gine -> nA
  tconv(nA, 49, 0, nB, nC);                // t1 (mean) -> nC [N x 64]
  tconv(nC, 59, 1, qb_, kb_);              // t2 (concat) -> kb_ [N x 256]
  gat(kb_, 34, nA, nB, nC, nB);            // gat -> nB = s

  // ================= merge =========================
  const size_t tot = (size_t)NN * 64 + (size_t)NE * 32;
  merge_kernel<<<(unsigned)((tot + 255) / 256), 256, 0, stream>>>(
      xo, nB, ea, w3, (float*)d_out);
}